// Net_12266426597866
// MI455X (gfx1250) — compile-verified
//
#include <hip/hip_runtime.h>
#include <hip/hip_bf16.h>

// ---------------------------------------------------------------------------
// MI455X (gfx1250) implementation of the spatio-temporal GNN reference.
//
// Key optimizations (reasoned from MI455X specs, no runtime available):
//  * The two Chebyshev graph-conv layers are linear => fully collapsed:
//      h2 = x·a0 + S(x)·(b0+a1) + S(S(x))·b1 + c1·S(1) + c0 + b2
//    This removes the 64-channel intermediate and cuts edge atomics from
//    T*E*64 (~8.2M) to ~T*E*4+E (~0.6M) — the prior dominant cost.
//  * The 16x16 temporal-conv channel mixes (tc2/tc3) are real GEMM shapes:
//    each wave owns 16 nodes and runs them on the matrix pipe as chained
//    V_WMMA_F32_16X16X4_F32 (4 per 16x16x16 product, 2 taps, 3+2 positions
//    = 40 WMMAs/wave), bias folded into the C accumulator.
//  * Rank-1 Q/K attention and rank-1 output head collapsed to per-head /
//    per-feature coefficients.
//  * Whole working set (~1.5MB) is L2-resident (192MB L2).
// ---------------------------------------------------------------------------

#define CN 2000
#define CE 32000
#define CT 4
#define CF 2
#define CH 8

typedef __attribute__((ext_vector_type(2))) float v2f;
typedef __attribute__((ext_vector_type(8))) float v8f;

static inline int cdiv(int a, int b) { return (a + b - 1) / b; }

// ---------------- zero fill ----------------
__global__ void zero_kernel(float* __restrict__ p, int n) {
    int i = blockIdx.x * blockDim.x + threadIdx.x;
    if (i < n) p[i] = 0.0f;
}

// ---------------- degree / norm ----------------
__global__ void deg_kernel(const int* __restrict__ src, const float* __restrict__ ew,
                           float* __restrict__ deg) {
    int e = blockIdx.x * blockDim.x + threadIdx.x;
    if (e < CE) atomicAdd(&deg[src[e]], ew[e]);
}

__global__ void invsqrt_kernel(const float* __restrict__ deg, float* __restrict__ isq) {
    int n = blockIdx.x * blockDim.x + threadIdx.x;
    if (n < CN) {
        float d = deg[n];
        isq[n] = (d > 0.0f) ? rsqrtf(d) : 0.0f;
    }
}

__global__ void norm_kernel(const int* __restrict__ src, const int* __restrict__ dst,
                            const float* __restrict__ ew, const float* __restrict__ isq,
                            float* __restrict__ norm) {
    int e = blockIdx.x * blockDim.x + threadIdx.x;
    if (e < CE) norm[e] = -isq[src[e]] * ew[e] * isq[dst[e]];
}

// ---------------- graph operator applications ----------------
// S1[n] = sum over edges with dst=n of norm[e]
__global__ void s1_kernel(const int* __restrict__ dst, const float* __restrict__ norm,
                          float* __restrict__ S1) {
    int e = blockIdx.x * blockDim.x + threadIdx.x;
    if (e < CE) atomicAdd(&S1[dst[e]], norm[e]);
}

// out[t][dst][f] += norm[e] * in[t][src][f]   (F=2)
__global__ void sop_kernel(const int* __restrict__ src, const int* __restrict__ dst,
                           const float* __restrict__ norm, const float* __restrict__ in,
                           float* __restrict__ out) {
    int i = blockIdx.x * blockDim.x + threadIdx.x;  // over T*E
    if (i >= CT * CE) return;
    int t = i / CE, e = i - t * CE;
    float nv = norm[e];
    int s = t * CN + src[e], d = t * CN + dst[e];
    atomicAdd(&out[d * 2 + 0], nv * in[s * 2 + 0]);
    atomicAdd(&out[d * 2 + 1], nv * in[s * 2 + 1]);
}

// ---------------- collapsed Chebyshev coefficients ----------------
// cc = [a0(2), b0(2), a1(2), b1(2), c0, c1]
// a0 = W0_1^T w0_2, b0 = W1_1^T w0_2, a1 = W0_1^T w1_2, b1 = W1_1^T w1_2
// c0 = b_1 . w0_2 ; c1 = b_1 . w1_2
__global__ void chebcoef_kernel(const float* __restrict__ c1w0, const float* __restrict__ c1w1,
                                const float* __restrict__ c1b, const float* __restrict__ c2w0,
                                const float* __restrict__ c2w1, float* __restrict__ cc) {
    int i = threadIdx.x;
    if (i < 8) {
        int which = i >> 1, f = i & 1;
        const float* W = (which & 1) ? c1w1 : c1w0;
        const float* v = (which < 2) ? c2w0 : c2w1;
        float s = 0.0f;
#pragma unroll 8
        for (int c = 0; c < 64; ++c) s += W[c * 2 + f] * v[c];
        cc[i] = s;
    } else if (i < 10) {
        const float* v = (i == 8) ? c2w0 : c2w1;
        float s = 0.0f;
#pragma unroll 8
        for (int c = 0; c < 64; ++c) s += c1b[c] * v[c];
        cc[i] = s;
    }
}

// h2[t][n] = x.a0 + Sx.(b0+a1) + SSx.b1 + c1*S1[n] + c0 + b2
__global__ void hcat_kernel(const float* __restrict__ x, const float* __restrict__ Sx,
                            const float* __restrict__ SSx, const float* __restrict__ S1,
                            const float* __restrict__ cc, const float* __restrict__ b2,
                            float* __restrict__ hcat) {
    int i = blockIdx.x * blockDim.x + threadIdx.x;  // T*N
    if (i >= CT * CN) return;
    int t = i / CN, n = i - t * CN;
    float h2 = cc[8] + b2[0] + cc[9] * S1[n]
             + x[i * 2 + 0] * cc[0] + x[i * 2 + 1] * cc[1]
             + Sx[i * 2 + 0] * (cc[2] + cc[4]) + Sx[i * 2 + 1] * (cc[3] + cc[5])
             + SSx[i * 2 + 0] * cc[6] + SSx[i * 2 + 1] * cc[7];
    hcat[n * CT + t] = h2;  // (n, t) layout for temporal conv
}

// ---------------- temporal conv chain on the matrix pipe ----------------
// Each wave owns 16 nodes. tc2/tc3 are (16ch x 16ch) mixes per tap =>
// 16x16x16 products = 4 chained V_WMMA_F32_16X16X4_F32 each.
// A (weights): row M = out-channel m holds K cols 4kc+2*half+{0,1}.
// B (activations): lane m (nodes) holds K rows 4kc+2*half+{0,1}.
// D/C: VGPR r = channel r+8*half, lane%16 = node.
__global__ __launch_bounds__(128) void tconv_wmma_kernel(
    const float* __restrict__ hcat,  // (N, T)
    const float* __restrict__ w1, const float* __restrict__ b1,
    const float* __restrict__ w2, const float* __restrict__ b2,
    const float* __restrict__ w3, const float* __restrict__ b3,
    const float* __restrict__ w4, const float* __restrict__ b4,
    float* __restrict__ h_st)
{
    const int lane = threadIdx.x & 31;
    const int wave = threadIdx.x >> 5;
    const int tile = blockIdx.x * 4 + wave;
    const int ntiles = CN / 16;  // 125
    if (tile >= ntiles) return;
    const int half = lane >> 4;
    const int m = lane & 15;
    const int node = tile * 16 + m;

    // padded temporal signal (both half-waves load the same node's values)
    float pd[6];
    pd[0] = 0.0f;
    pd[5] = 0.0f;
#pragma unroll
    for (int t = 0; t < 4; ++t) pd[t + 1] = hcat[node * CT + t];

    // ---- tc1 (width 3) computed directly in B-operand channel layout ----
    // lane's 8 channels: c_i = 4*(i>>1) + 2*half + (i&1)
    float y1b[8][4];
#pragma unroll
    for (int i = 0; i < 8; ++i) {
        int c = 4 * (i >> 1) + 2 * half + (i & 1);
        float w10 = w1[c * 3 + 0], w11 = w1[c * 3 + 1], w12 = w1[c * 3 + 2];
        float bb = b1[c];
#pragma unroll
        for (int p = 0; p < 4; ++p)
            y1b[i][p] = bb + w10 * pd[p] + w11 * pd[p + 1] + w12 * pd[p + 2];
    }

    // ---- A operands (weights) for tc2 / tc3 ----
    v2f a2[2][4], a3[2][4];
#pragma unroll
    for (int k = 0; k < 2; ++k)
#pragma unroll
        for (int kc = 0; kc < 4; ++kc) {
            int k0 = 4 * kc + 2 * half;
            v2f t;
            t.x = w2[m * 32 + k0 * 2 + k];
            t.y = w2[m * 32 + (k0 + 1) * 2 + k];
            a2[k][kc] = t;
            t.x = w3[m * 32 + k0 * 2 + k];
            t.y = w3[m * 32 + (k0 + 1) * 2 + k];
            a3[k][kc] = t;
        }

    // ---- tc2 (width 2): 3 output positions, 8 WMMAs each ----
    float y2d[3][8];
#pragma unroll
    for (int p = 0; p < 3; ++p) {
        v8f c;
#pragma unroll
        for (int r = 0; r < 8; ++r) c[r] = b2[r + 8 * half];
#pragma unroll
        for (int k = 0; k < 2; ++k)
#pragma unroll
            for (int kc = 0; kc < 4; ++kc) {
                v2f bv;
                bv.x = y1b[2 * kc + 0][p + k];
                bv.y = y1b[2 * kc + 1][p + k];
                c = __builtin_amdgcn_wmma_f32_16x16x4_f32(
                        false, a2[k][kc], false, bv, (short)0, c, false, false);
            }
#pragma unroll
        for (int r = 0; r < 8; ++r) y2d[p][r] = c[r];
    }

    // ---- D-layout -> B-layout: exchange halves, compile-time reg indices ----
    float y2b[8][3];
#pragma unroll
    for (int p = 0; p < 3; ++p) {
        float oth[8];
#pragma unroll
        for (int r = 0; r < 8; ++r) oth[r] = __shfl_xor(y2d[p][r], 16, 32);
#pragma unroll
        for (int kc = 0; kc < 4; ++kc)
#pragma unroll
            for (int j = 0; j < 2; ++j) {
                const int cL = 4 * kc + j;      // channel if lower half
                const int cU = 4 * kc + 2 + j;  // channel if upper half
                float vL = (cL < 8) ? y2d[p][cL] : oth[cL - 8];
                float vU = (cU < 8) ? oth[cU] : y2d[p][cU - 8];
                y2b[2 * kc + j][p] = half ? vU : vL;
            }
    }

    // ---- tc3 (width 2): 2 output positions, 8 WMMAs each ----
    float y3d[2][8];
#pragma unroll
    for (int p = 0; p < 2; ++p) {
        v8f c;
#pragma unroll
        for (int r = 0; r < 8; ++r) c[r] = b3[r + 8 * half];
#pragma unroll
        for (int k = 0; k < 2; ++k)
#pragma unroll
            for (int kc = 0; kc < 4; ++kc) {
                v2f bv;
                bv.x = y2b[2 * kc + 0][p + k];
                bv.y = y2b[2 * kc + 1][p + k];
                c = __builtin_amdgcn_wmma_f32_16x16x4_f32(
                        false, a3[k][kc], false, bv, (short)0, c, false, false);
            }
#pragma unroll
        for (int r = 0; r < 8; ++r) y3d[p][r] = c[r];
    }

    // ---- tc4: 16 -> 1 dot, cross-half reduce ----
    float s = 0.0f;
#pragma unroll
    for (int r = 0; r < 8; ++r) {
        int ch = r + 8 * half;
        s += w4[ch * 2 + 0] * y3d[0][r] + w4[ch * 2 + 1] * y3d[1][r];
    }
    s += __shfl_xor(s, 16, 32);
    if (half == 0) h_st[node] = b4[0] + s;
}

// ---------------- rank-1 Q/K coefficient collapse ----------------
__global__ void qkcoef_kernel(const float* __restrict__ qw, const float* __restrict__ qb,
                              const float* __restrict__ kw, const float* __restrict__ kb,
                              float* __restrict__ coef) {
    int i = threadIdx.x;  // 32 threads: (head, which)
    if (i >= CH * 4) return;
    int h = i >> 2, which = i & 3;
    const float* aa = (which < 2) ? qw : qb;
    const float* bb = (which & 1) ? kb : kw;
    float s = 0.0f;
#pragma unroll
    for (int dk = 0; dk < 16; ++dk) s += aa[h * 16 + dk] * bb[h * 16 + dk];
    coef[h * 4 + which] = s;
}

__global__ void prods_kernel(const int* __restrict__ src, const int* __restrict__ dst,
                             const float* __restrict__ h_st, const float* __restrict__ coef,
                             float* __restrict__ prods) {
    int i = blockIdx.x * blockDim.x + threadIdx.x;  // E*H
    if (i >= CE * CH) return;
    int e = i >> 3, h = i & 7;
    float hs = h_st[src[e]], hd = h_st[dst[e]];
    const float* c = &coef[h * 4];
    prods[i] = 0.25f * (c[0] * hs * hd + c[1] * hs + c[2] * hd + c[3]);
}

// ---------------- softmax over all E edges, per head ----------------
__global__ void softmax_reduce_kernel(const float* __restrict__ prods,
                                      float* __restrict__ smax, float* __restrict__ ssum) {
    const int h = blockIdx.x;  // one block per head
    __shared__ float red[256];
    float m = -3.402823466e38f;
    for (int e = threadIdx.x; e < CE; e += 256) m = fmaxf(m, prods[e * CH + h]);
    red[threadIdx.x] = m;
    __syncthreads();
    for (int s = 128; s > 0; s >>= 1) {
        if (threadIdx.x < s) red[threadIdx.x] = fmaxf(red[threadIdx.x], red[threadIdx.x + s]);
        __syncthreads();
    }
    m = red[0];
    __syncthreads();
    float sum = 0.0f;
    for (int e = threadIdx.x; e < CE; e += 256) sum += __expf(prods[e * CH + h] - m);
    red[threadIdx.x] = sum;
    __syncthreads();
    for (int s = 128; s > 0; s >>= 1) {
        if (threadIdx.x < s) red[threadIdx.x] += red[threadIdx.x + s];
        __syncthreads();
    }
    if (threadIdx.x == 0) { smax[h] = m; ssum[h] = red[0]; }
}

// gx only needed for edge ids j < N (div_op columns are node ids used as edge ids)
__global__ void gx_kernel(const int* __restrict__ src, const int* __restrict__ dst,
                          const float* __restrict__ prods, const float* __restrict__ smax,
                          const float* __restrict__ ssum, const float* __restrict__ h_st,
                          float* __restrict__ gx) {
    int i = blockIdx.x * blockDim.x + threadIdx.x;  // N*H
    if (i >= CN * CH) return;
    int j = i >> 3, h = i & 7;
    float A = __expf(prods[j * CH + h] - smax[h]) / ssum[h];
    gx[i] = A * (h_st[src[j]] - h_st[dst[j]]);
}

// ggx[n,h] = sum over edges e with src[e]=n of edge_attr[dst[e]] * gx[dst[e], h]
__global__ void ggx_kernel(const int* __restrict__ src, const int* __restrict__ dst,
                           const float* __restrict__ ea, const float* __restrict__ gx,
                           float* __restrict__ ggx) {
    int i = blockIdx.x * blockDim.x + threadIdx.x;  // E*H
    if (i >= CE * CH) return;
    int e = i >> 3, h = i & 7;
    int d = dst[e];
    atomicAdd(&ggx[src[e] * CH + h], ea[d] * gx[d * CH + h]);
}

// ---------------- rank-1 output head collapse ----------------
__global__ void woutcoef_kernel(const float* __restrict__ w1w, const float* __restrict__ w1b,
                                const float* __restrict__ w2w, const float* __restrict__ w2b,
                                float* __restrict__ acoef, float* __restrict__ bcoef) {
    int f = threadIdx.x;
    if (f >= CF) return;
    float a = 0.0f, b = 0.0f;
#pragma unroll 8
    for (int c = 0; c < 64; ++c) {
        a += w1w[c] * w2w[f * 64 + c];
        b += w1b[c] * w2w[f * 64 + c];
    }
    acoef[f] = a;
    bcoef[f] = b + w2b[f];
}

__global__ void copy_kernel(const float* __restrict__ x, float* __restrict__ out) {
    int i = blockIdx.x * blockDim.x + threadIdx.x;  // 3*N*F
    if (i < 3 * CN * CF) out[i] = x[CN * CF + i];
}

__global__ void final_kernel(const float* __restrict__ x, const float* __restrict__ ggx,
                             const float* __restrict__ acoef, const float* __restrict__ bcoef,
                             float* __restrict__ out) {
    int n = blockIdx.x * blockDim.x + threadIdx.x;
    if (n >= CN) return;
    float m = 0.0f;
#pragma unroll
    for (int h = 0; h < CH; ++h) m += ggx[n * CH + h];
    m *= 0.125f;
#pragma unroll
    for (int f = 0; f < CF; ++f)
        out[3 * CN * CF + n * CF + f] =
            x[3 * CN * CF + n * CF + f] + m * acoef[f] + bcoef[f];
}

// ---------------------------------------------------------------------------
extern "C" void kernel_launch(void* const* d_in, const int* in_sizes, int n_in,
                              void* d_out, int out_size, void* d_ws, size_t ws_size,
                              hipStream_t stream) {
    (void)in_sizes; (void)n_in; (void)out_size; (void)ws_size;

    const float* x_list = (const float*)d_in[1];
    const int*   ei     = (const int*)d_in[2];
    const int*   src    = ei;
    const int*   dst    = ei + CE;
    const float* ea     = (const float*)d_in[3];
    const float* c1w0   = (const float*)d_in[4];
    const float* c1w1   = (const float*)d_in[5];
    const float* c1b    = (const float*)d_in[6];
    const float* c2w0   = (const float*)d_in[7];
    const float* c2w1   = (const float*)d_in[8];
    const float* c2b    = (const float*)d_in[9];
    const float* t1w    = (const float*)d_in[10];
    const float* t1b    = (const float*)d_in[11];
    const float* t2w    = (const float*)d_in[12];
    const float* t2b    = (const float*)d_in[13];
    const float* t3w    = (const float*)d_in[14];
    const float* t3b    = (const float*)d_in[15];
    const float* t4w    = (const float*)d_in[16];
    const float* t4b    = (const float*)d_in[17];
    const float* qw     = (const float*)d_in[18];
    const float* qb     = (const float*)d_in[19];
    const float* kw     = (const float*)d_in[20];
    const float* kb     = (const float*)d_in[21];
    const float* w1w    = (const float*)d_in[22];
    const float* w1b    = (const float*)d_in[23];
    const float* w2w    = (const float*)d_in[24];
    const float* w2b    = (const float*)d_in[25];
    float* out = (float*)d_out;

    // workspace layout (floats); zero-initialized prefix first
    float* ws  = (float*)d_ws;
    float* deg = ws;                    // N      [zero]
    float* Sx  = deg + CN;              // T*N*2  [zero]
    float* SSx = Sx + CT * CN * 2;      // T*N*2  [zero]
    float* S1  = SSx + CT * CN * 2;     // N      [zero]
    float* ggx = S1 + CN;               // N*H    [zero]
    int zcount = CN + CT * CN * 2 + CT * CN * 2 + CN + CN * CH;
    float* isq   = ggx + CN * CH;       // N
    float* norm  = isq + CN;            // E
    float* cc    = norm + CE;           // 10
    float* hcat  = cc + 10;             // N*T
    float* h_st  = hcat + CN * CT;      // N
    float* coef  = h_st + CN;           // H*4
    float* prods = coef + CH * 4;       // E*H
    float* smax  = prods + CE * CH;     // H
    float* ssum  = smax + CH;           // H
    float* gx    = ssum + CH;           // N*H
    float* acoef = gx + CN * CH;        // F
    float* bcoef = acoef + CF;          // F

    zero_kernel<<<cdiv(zcount, 256), 256, 0, stream>>>(ws, zcount);

    deg_kernel<<<cdiv(CE, 256), 256, 0, stream>>>(src, ea, deg);
    invsqrt_kernel<<<cdiv(CN, 256), 256, 0, stream>>>(deg, isq);
    norm_kernel<<<cdiv(CE, 256), 256, 0, stream>>>(src, dst, ea, isq, norm);

    s1_kernel<<<cdiv(CE, 256), 256, 0, stream>>>(dst, norm, S1);
    sop_kernel<<<cdiv(CT * CE, 256), 256, 0, stream>>>(src, dst, norm, x_list, Sx);
    sop_kernel<<<cdiv(CT * CE, 256), 256, 0, stream>>>(src, dst, norm, Sx, SSx);

    chebcoef_kernel<<<1, 32, 0, stream>>>(c1w0, c1w1, c1b, c2w0, c2w1, cc);
    hcat_kernel<<<cdiv(CT * CN, 256), 256, 0, stream>>>(x_list, Sx, SSx, S1, cc, c2b, hcat);

    tconv_wmma_kernel<<<cdiv(CN / 16, 4), 128, 0, stream>>>(
        hcat, t1w, t1b, t2w, t2b, t3w, t3b, t4w, t4b, h_st);

    qkcoef_kernel<<<1, 32, 0, stream>>>(qw, qb, kw, kb, coef);
    prods_kernel<<<cdiv(CE * CH, 256), 256, 0, stream>>>(src, dst, h_st, coef, prods);
    softmax_reduce_kernel<<<CH, 256, 0, stream>>>(prods, smax, ssum);
    gx_kernel<<<cdiv(CN * CH, 256), 256, 0, stream>>>(src, dst, prods, smax, ssum, h_st, gx);
    ggx_kernel<<<cdiv(CE * CH, 256), 256, 0, stream>>>(src, dst, ea, gx, ggx);

    woutcoef_kernel<<<1, 64, 0, stream>>>(w1w, w1b, w2w, w2b, acoef, bcoef);
    copy_kernel<<<cdiv(3 * CN * CF, 256), 256, 0, stream>>>(x_list, out);
    final_kernel<<<cdiv(CN, 256), 256, 0, stream>>>(x_list, ggx, acoef, bcoef, out);
}